// TGraphConvolution_10574209483501
// MI455X (gfx1250) — compile-verified
//
#include <hip/hip_runtime.h>

// ---------------------------------------------------------------------------
// TGraphConvolution for MI455X (gfx1250, wave32)
//   support = (X @ W) * t[:,None]        -> V_WMMA_F32_16X16X4_F32 GEMM
//   msgs    = support[src] * ev[:,None]  -> float4 gather (L2-resident)
//   out     = segment_sum(msgs, dst)+b   -> global_atomic_add_f32 into L2
// Memory-bound: ~450MB total traffic @ 23.3TB/s; GEMM uses the matrix pipe.
// ---------------------------------------------------------------------------

typedef __attribute__((ext_vector_type(2))) float v2f;
typedef __attribute__((ext_vector_type(8))) float v8f;

#define N_NODES 50000
#define F_IN    128
#define F_OUT   64

// Kernel 1: support[N,64] = (X[N,128] @ W[128,64]) * t[:,None]
// 256 threads = 8 waves; each wave computes one 16x16 tile.
// Block covers 32 rows x 64 cols (2 row-tiles x 4 col-tiles).
__global__ __launch_bounds__(256) void gemm_wmma_f32(
    const float* __restrict__ x, const float* __restrict__ w,
    const float* __restrict__ t, float* __restrict__ support)
{
    const int lane    = threadIdx.x & 31;
    const int wave    = threadIdx.x >> 5;
    const int rowBase = (blockIdx.x * 2 + (wave >> 2)) * 16;  // wave-uniform
    const int colBase = (wave & 3) * 16;
    if (rowBase >= N_NODES) return;   // uniform branch: EXEC stays all-1s

    const int mn = lane & 15;   // M for A / N for B,C,D
    const int hi = lane >> 4;   // 0: K=k0,k0+1   1: K=k0+2,k0+3

    // Clamp A-row for loads so EXEC stays full through the WMMAs.
    int arow = rowBase + mn;
    if (arow >= N_NODES) arow = N_NODES - 1;
    const float* __restrict__ xrow = x + (size_t)arow * F_IN + 2 * hi;
    const float* __restrict__ wcol = w + colBase + mn;

    v8f c = {};
#pragma unroll
    for (int k0 = 0; k0 < F_IN; k0 += 4) {
        v2f a, b;
        // A 16x4 fp32 fragment (2 VGPRs):
        //   lane<16 : A[M=mn][k0+0], A[M=mn][k0+1]
        //   lane>=16: A[M=mn][k0+2], A[M=mn][k0+3]
        a.x = xrow[k0];
        a.y = xrow[k0 + 1];
        // B 4x16 fp32 fragment (2 VGPRs), W is row-major [128,64]:
        //   lane<16 : B[K=k0+0][N=mn], B[K=k0+1][N=mn]
        //   lane>=16: B[K=k0+2][N=mn], B[K=k0+3][N=mn]
        b.x = wcol[(size_t)(k0 + 2 * hi) * F_OUT];
        b.y = wcol[(size_t)(k0 + 2 * hi + 1) * F_OUT];
        // 8-arg form: (neg_a, A, neg_b, B, c_mod, C, reuse_a, reuse_b)
        c = __builtin_amdgcn_wmma_f32_16x16x4_f32(false, a, false, b,
                                                  (short)0, c, false, false);
    }

    // C/D layout: VGPR i holds M = i + 8*hi, N = mn. Fuse t[row] scale.
#pragma unroll
    for (int i = 0; i < 8; ++i) {
        const int row = rowBase + 8 * hi + i;
        if (row < N_NODES)
            support[(size_t)row * F_OUT + colBase + mn] = c[i] * t[row];
    }
}

// Kernel 2: out[n][f] = bias[f]  (d_out is poisoned by the harness)
__global__ __launch_bounds__(256) void init_bias(
    const float* __restrict__ bias, float* __restrict__ out)
{
    const int idx = blockIdx.x * 256 + threadIdx.x;
    if (idx < N_NODES * F_OUT) out[idx] = bias[idx & (F_OUT - 1)];
}

// Kernel 3: out[dst[e]] += support[src[e]] * ev[e]
// 16 lanes per edge, float4 per lane -> coalesced 256B row gather,
// atomics coalesce/combine in the 192MB L2.
__global__ __launch_bounds__(256) void scatter_edges(
    const float* __restrict__ support, const int* __restrict__ src,
    const int* __restrict__ dst, const float* __restrict__ ev,
    float* __restrict__ out, int E)
{
    const int tid = blockIdx.x * 256 + threadIdx.x;
    const int e   = tid >> 4;
    if (e >= E) return;
    const int f = (tid & 15) * 4;

    const int   s = src[e];
    const int   d = dst[e];
    const float v = ev[e];

    const float4 m = *(const float4*)(support + (size_t)s * F_OUT + f);
    float* op = out + (size_t)d * F_OUT + f;
    atomicAdd(op + 0, m.x * v);
    atomicAdd(op + 1, m.y * v);
    atomicAdd(op + 2, m.z * v);
    atomicAdd(op + 3, m.w * v);
}

extern "C" void kernel_launch(void* const* d_in, const int* in_sizes, int n_in,
                              void* d_out, int out_size, void* d_ws, size_t ws_size,
                              hipStream_t stream)
{
    const float* x    = (const float*)d_in[0];   // [N,128]
    const float* t    = (const float*)d_in[1];   // [N]
    const int*   src  = (const int*)  d_in[2];   // [E]
    const int*   dst  = (const int*)  d_in[3];   // [E]
    const float* ev   = (const float*)d_in[4];   // [E]
    const float* w    = (const float*)d_in[5];   // [128,64]
    const float* bias = (const float*)d_in[6];   // [64]
    float*       out  = (float*)d_out;           // [N,64]
    float*       supp = (float*)d_ws;            // [N,64] scratch (12.8 MB)
    const int E = in_sizes[2];

    // K1: WMMA GEMM + t-scale -> d_ws
    {
        const int blocks = (N_NODES + 31) / 32;  // 32 rows per block
        gemm_wmma_f32<<<blocks, 256, 0, stream>>>(x, w, t, supp);
    }
    // K2: out = bias (broadcast)
    {
        const int total  = N_NODES * F_OUT;
        const int blocks = (total + 255) / 256;
        init_bias<<<blocks, 256, 0, stream>>>(bias, out);
    }
    // K3: edge gather/scale/scatter-add
    {
        const long long work = (long long)E * 16;
        const int blocks = (int)((work + 255) / 256);
        scatter_edges<<<blocks, 256, 0, stream>>>(supp, src, dst, ev, out, E);
    }
}